// GAT_28286654612182
// MI455X (gfx1250) — compile-verified
//
#include <hip/hip_runtime.h>
#include <hip/hip_bf16.h>

typedef __attribute__((ext_vector_type(2))) float v2f;
typedef __attribute__((ext_vector_type(8))) float v8f;

#define SLOPE 0.2f

__device__ __forceinline__ float leaky(float x) { return x > 0.0f ? x : SLOPE * x; }

// float atomicMax via int punning (valid for finite values)
__device__ __forceinline__ void atomicMaxF(float* addr, float val) {
    if (val >= 0.0f)
        atomicMax((int*)addr, __float_as_int(val));
    else
        atomicMin((unsigned int*)addr, __float_as_uint(val));
}

#define WMMA_F32(a, b, c) \
    __builtin_amdgcn_wmma_f32_16x16x4_f32(false, (a), false, (b), (short)0, (c), false, false)

// ---------------------------------------------------------------------------
// Tiled fp32 WMMA GEMM: D[M,N] = act(A[M,K] @ B[K,N] + bias)
// One wave computes a 16x64 stripe (4 WMMA tiles) so the A fragment loaded
// each K-step is reused by 4 V_WMMA_F32_16X16X4_F32 ops (loads/WMMA: 4->2.5).
// Requirements (met by all call sites): M%16==0, K%4==0, N%256==0.
// block = 128 threads (4 waves, 4 adjacent stripes); grid = (M/16, N/256).
// ---------------------------------------------------------------------------
__global__ void gat_gemm_wmma_f32(const float* __restrict__ A, const float* __restrict__ B,
                                  const float* __restrict__ bias, float* __restrict__ D,
                                  int M, int K, int Nout, int relu) {
    const int wave = threadIdx.x >> 5;
    const int lane = threadIdx.x & 31;
    const int row0 = blockIdx.x * 16;
    const int col0 = (blockIdx.y * 4 + wave) * 64;
    if (col0 >= Nout) return;  // uniform per wave

    const int lm = lane & 15;        // M (for A) / N (for B,C,D) within tile
    const int lk = (lane >> 4) * 2;  // K pair base: 0 or 2

    const float* __restrict__ Arow = A + (size_t)(row0 + lm) * K + lk;
    const float* __restrict__ Bb = B + (size_t)lk * Nout + (col0 + lm);

    v8f acc0 = {}, acc1 = {}, acc2 = {}, acc3 = {};
    for (int k = 0; k < K; k += 4) {
        v2f a;
        a[0] = Arow[k + 0];
        a[1] = Arow[k + 1];
        const float* __restrict__ Bk0 = Bb + (size_t)k * Nout;
        const float* __restrict__ Bk1 = Bk0 + Nout;
        v2f b0, b1, b2, b3;
        b0[0] = Bk0[0];  b0[1] = Bk1[0];
        b1[0] = Bk0[16]; b1[1] = Bk1[16];
        b2[0] = Bk0[32]; b2[1] = Bk1[32];
        b3[0] = Bk0[48]; b3[1] = Bk1[48];
        acc0 = WMMA_F32(a, b0, acc0);
        acc1 = WMMA_F32(a, b1, acc1);
        acc2 = WMMA_F32(a, b2, acc2);
        acc3 = WMMA_F32(a, b3, acc3);
    }

    const int mbase = row0 + ((lane >> 4) << 3);  // lanes 16-31 hold rows M+8..M+15
    v8f accs[4] = {acc0, acc1, acc2, acc3};
#pragma unroll
    for (int t = 0; t < 4; ++t) {
        const int n = col0 + t * 16 + lm;
        const float bv = bias ? bias[n] : 0.0f;
#pragma unroll
        for (int v = 0; v < 8; ++v) {
            float val = accs[t][v] + bv;
            if (relu) val = fmaxf(val, 0.0f);
            D[(size_t)(mbase + v) * Nout + n] = val;
        }
    }
}

// ---------------------------------------------------------------------------
// Per-(node,head) attention coefficients: as/ad[n,h] = <h[n,h,:], a_src/dst[h,:]>
// ---------------------------------------------------------------------------
__global__ void gat_alpha(const float* __restrict__ h, const float* __restrict__ a_src,
                          const float* __restrict__ a_dst, float* __restrict__ as,
                          float* __restrict__ ad, int Nn, int H, int C) {
    int i = blockIdx.x * blockDim.x + threadIdx.x;
    if (i >= Nn * H) return;
    const int nd = i / H, hh = i % H;
    const float* __restrict__ row = h + (size_t)nd * H * C + (size_t)hh * C;
    const float* __restrict__ vs = a_src + (size_t)hh * C;
    const float* __restrict__ vd = a_dst + (size_t)hh * C;
    float s1 = 0.0f, s2 = 0.0f;
    for (int c = 0; c < C; ++c) {
        float v = row[c];
        s1 += v * vs[c];
        s2 += v * vd[c];
    }
    as[i] = s1;
    ad[i] = s2;
}

// emax[n,h] initialized with the (always-valid) self-loop logit
__global__ void gat_emax_init(const float* __restrict__ as, const float* __restrict__ ad,
                              float* __restrict__ emax, int NH) {
    int i = blockIdx.x * blockDim.x + threadIdx.x;
    if (i >= NH) return;
    emax[i] = leaky(as[i] + ad[i]);
}

// per-edge max of attention logits into emax[dst,h]
__global__ void gat_edge_max(const int* __restrict__ src, const int* __restrict__ dst,
                             const float* __restrict__ as, const float* __restrict__ ad,
                             float* __restrict__ emax, int E, int H) {
    int i = blockIdx.x * blockDim.x + threadIdx.x;
    if (i >= E) return;
    const int s = src[i], d = dst[i];
    if (s == d) return;  // removed self-loop (invalid edge)
    for (int hh = 0; hh < H; ++hh) {
        float e = leaky(as[s * H + hh] + ad[d * H + hh]);
        atomicMaxF(&emax[d * H + hh], e);
    }
}

__global__ void gat_zero(float* __restrict__ p, size_t n) {
    size_t i = (size_t)blockIdx.x * blockDim.x + threadIdx.x;
    if (i < n) p[i] = 0.0f;
}

// ---------------------------------------------------------------------------
// Wave-per-edge weighted scatter. Edges [0,E) are real (skip s==d);
// edges [E,E+Nn) are self-loops. Lanes stripe the H*C channels (coalesced
// gather of h[src], atomic f32 scatter into out[dst]). Lanes < H accumulate
// the softmax denominator.
// ---------------------------------------------------------------------------
__global__ void gat_edge_aggregate(const int* __restrict__ src, const int* __restrict__ dst,
                                   const float* __restrict__ as, const float* __restrict__ ad,
                                   const float* __restrict__ emax,
                                   const float* __restrict__ hfeat,
                                   float* __restrict__ outacc, float* __restrict__ denom,
                                   int E, int Nn, int H, int C) {
    const int widx = (int)(((size_t)blockIdx.x * blockDim.x + threadIdx.x) >> 5);
    const int lane = threadIdx.x & 31;
    const int total = E + Nn;
    if (widx >= total) return;

    int s, d;
    if (widx < E) {
        s = src[widx];
        d = dst[widx];
        if (s == d) return;  // invalid edge
    } else {
        s = d = widx - E;    // self-loop
    }

    float w[4];  // H <= 4 at every call site
    for (int hh = 0; hh < H; ++hh) {
        float e = leaky(as[s * H + hh] + ad[d * H + hh]);
        w[hh] = __expf(e - emax[d * H + hh]);
    }
    if (lane < H) atomicAdd(&denom[d * H + lane], w[lane]);

    const float* __restrict__ hs = hfeat + (size_t)s * H * C;
    float* __restrict__ od = outacc + (size_t)d * H * C;
    for (int hh = 0; hh < H; ++hh) {
        const float whh = w[hh];
        const int cend = (hh + 1) * C;
        for (int c = hh * C + lane; c < cend; c += 32) {
            atomicAdd(&od[c], whh * hs[c]);
        }
    }
}

// out[n,c] = relu(out[n,c] / denom[n, c>>cshift] + bias[c])
__global__ void gat_normalize(float* __restrict__ out, const float* __restrict__ denom,
                              const float* __restrict__ bias, int Nn, int H, int HC,
                              int cshift) {
    size_t i = (size_t)blockIdx.x * blockDim.x + threadIdx.x;
    if (i >= (size_t)Nn * HC) return;
    const int nd = (int)(i / HC);
    const int c = (int)(i % HC);
    const int hh = c >> cshift;
    float v = out[i] / denom[nd * H + hh] + bias[c];
    out[i] = fmaxf(v, 0.0f);
}

// ---------------------------------------------------------------------------
// Final head: logits = relu(m2 @ lw3 + lb3); out = softmax(logits) per row.
// One thread per node; lw3 (40 KB) stays cache-resident.
// ---------------------------------------------------------------------------
__global__ void gat_head(const float* __restrict__ m2, const float* __restrict__ lw3,
                         const float* __restrict__ lb3, float* __restrict__ out, int Nn) {
    int nd = blockIdx.x * blockDim.x + threadIdx.x;
    if (nd >= Nn) return;
    const float* __restrict__ row = m2 + (size_t)nd * 1024;
    float lg[10];
#pragma unroll
    for (int j = 0; j < 10; ++j) lg[j] = lb3[j];
    for (int k = 0; k < 1024; ++k) {
        const float xv = row[k];
        const float* __restrict__ wrow = lw3 + (size_t)k * 10;
#pragma unroll
        for (int j = 0; j < 10; ++j) lg[j] += xv * wrow[j];
    }
    float mx = -1e30f;
#pragma unroll
    for (int j = 0; j < 10; ++j) {
        lg[j] = fmaxf(lg[j], 0.0f);
        mx = fmaxf(mx, lg[j]);
    }
    float sum = 0.0f;
#pragma unroll
    for (int j = 0; j < 10; ++j) {
        lg[j] = __expf(lg[j] - mx);
        sum += lg[j];
    }
    const float inv = 1.0f / sum;
#pragma unroll
    for (int j = 0; j < 10; ++j) out[(size_t)nd * 10 + j] = lg[j] * inv;
}

// ---------------------------------------------------------------------------

static inline int cdiv(long long a, long long b) { return (int)((a + b - 1) / b); }

extern "C" void kernel_launch(void* const* d_in, const int* in_sizes, int n_in,
                              void* d_out, int out_size, void* d_ws, size_t ws_size,
                              hipStream_t stream) {
    const float* x      = (const float*)d_in[0];
    const int*   ei     = (const int*)d_in[1];
    const float* W0     = (const float*)d_in[2];
    const float* a_src0 = (const float*)d_in[3];
    const float* a_dst0 = (const float*)d_in[4];
    const float* b0     = (const float*)d_in[5];
    const float* W1     = (const float*)d_in[6];
    const float* a_src1 = (const float*)d_in[7];
    const float* a_dst1 = (const float*)d_in[8];
    const float* b1     = (const float*)d_in[9];
    const float* lw1    = (const float*)d_in[10];
    const float* lb1    = (const float*)d_in[11];
    const float* lw2    = (const float*)d_in[12];
    const float* lb2    = (const float*)d_in[13];
    const float* lw3    = (const float*)d_in[14];
    const float* lb3    = (const float*)d_in[15];
    float* out = (float*)d_out;

    const int N = in_sizes[0] / 128;  // 50000 (multiple of 16)
    const int E = in_sizes[1] / 2;    // 400000
    const int* src = ei;
    const int* dst = ei + E;

    // ---- workspace layout (floats) -----------------------------------------
    float* ws = (float*)d_ws;
    const size_t Ns = (size_t)N;
    float* h0   = ws;                 // N*256
    float* out0 = ws + Ns * 256;      // N*256
    float* h1   = ws + Ns * 512;      // N*256
    float* out1 = ws + Ns * 768;      // N*256
    float* m2   = ws;                 // N*1024 (reuses h0..out1 after m1 done)
    float* m1   = ws + Ns * 1024;     // N*512
    float* sm   = ws + Ns * 1536;     // small arrays:
    float* as0 = sm;                  // N*4
    float* ad0 = sm + Ns * 4;         // N*4
    float* em0 = sm + Ns * 8;         // N*4
    float* dn0 = sm + Ns * 12;        // N*4
    float* as1 = sm + Ns * 16;        // N
    float* ad1 = sm + Ns * 17;        // N
    float* em1 = sm + Ns * 18;        // N
    float* dn1 = sm + Ns * 19;        // N

    const dim3 blk128(128), blk256(256);
    const int nWaveBlocks = cdiv((long long)(E + N) * 32, 256);

    // ================= Layer 0: GATConv 128 -> 4x64 =========================
    // h0 = x @ W0
    gat_gemm_wmma_f32<<<dim3(N / 16, 256 / 256), blk128, 0, stream>>>(
        x, W0, nullptr, h0, N, 128, 256, 0);
    gat_alpha<<<cdiv((long long)N * 4, 256), blk256, 0, stream>>>(
        h0, a_src0, a_dst0, as0, ad0, N, 4, 64);
    gat_emax_init<<<cdiv((long long)N * 4, 256), blk256, 0, stream>>>(as0, ad0, em0, N * 4);
    gat_edge_max<<<cdiv(E, 256), blk256, 0, stream>>>(src, dst, as0, ad0, em0, E, 4);
    gat_zero<<<cdiv((long long)N * 256, 256), blk256, 0, stream>>>(out0, Ns * 256);
    gat_zero<<<cdiv((long long)N * 4, 256), blk256, 0, stream>>>(dn0, Ns * 4);
    gat_edge_aggregate<<<nWaveBlocks, blk256, 0, stream>>>(
        src, dst, as0, ad0, em0, h0, out0, dn0, E, N, 4, 64);
    gat_normalize<<<cdiv((long long)N * 256, 256), blk256, 0, stream>>>(
        out0, dn0, b0, N, 4, 256, 6);

    // ================= Layer 1: GATConv 256 -> 1x256 ========================
    gat_gemm_wmma_f32<<<dim3(N / 16, 256 / 256), blk128, 0, stream>>>(
        out0, W1, nullptr, h1, N, 256, 256, 0);
    gat_alpha<<<cdiv(N, 256), blk256, 0, stream>>>(h1, a_src1, a_dst1, as1, ad1, N, 1, 256);
    gat_emax_init<<<cdiv(N, 256), blk256, 0, stream>>>(as1, ad1, em1, N);
    gat_edge_max<<<cdiv(E, 256), blk256, 0, stream>>>(src, dst, as1, ad1, em1, E, 1);
    gat_zero<<<cdiv((long long)N * 256, 256), blk256, 0, stream>>>(out1, Ns * 256);
    gat_zero<<<cdiv(N, 256), blk256, 0, stream>>>(dn1, Ns);
    gat_edge_aggregate<<<nWaveBlocks, blk256, 0, stream>>>(
        src, dst, as1, ad1, em1, h1, out1, dn1, E, N, 1, 256);
    gat_normalize<<<cdiv((long long)N * 256, 256), blk256, 0, stream>>>(
        out1, dn1, b1, N, 1, 256, 8);

    // ================= MLP head =============================================
    // m1 = relu(out1 @ lw1 + lb1)   [N,512]
    gat_gemm_wmma_f32<<<dim3(N / 16, 512 / 256), blk128, 0, stream>>>(
        out1, lw1, lb1, m1, N, 256, 512, 1);
    // m2 = relu(m1 @ lw2 + lb2)     [N,1024]  (reuses h0..out1 region)
    gat_gemm_wmma_f32<<<dim3(N / 16, 1024 / 256), blk128, 0, stream>>>(
        m1, lw2, lb2, m2, N, 512, 1024, 1);
    // logits + softmax -> d_out     [N,10]
    gat_head<<<cdiv(N, 256), blk256, 0, stream>>>(m2, lw3, lb3, out, N);
}